// LSTM_12017318494801
// MI455X (gfx1250) — compile-verified
//
#include <hip/hip_runtime.h>
#include <cstdint>

// Problem constants (fixed by the reference).
static constexpr int    B_  = 2048;
static constexpr int    T_  = 2048;
static constexpr size_t NBT = (size_t)B_ * (size_t)T_;
// Output layout: [loss(1) | imputations(B*T) | is_train(B) | evals(B*T) | eval_masks(B*T)]
static constexpr size_t OFF_IMP   = 1;
static constexpr size_t OFF_TRAIN = OFF_IMP + NBT;
static constexpr size_t OFF_EVALS = OFF_TRAIN + (size_t)B_;
static constexpr size_t OFF_EM    = OFF_EVALS + NBT;

typedef __attribute__((ext_vector_type(2))) float v2f;
typedef __attribute__((ext_vector_type(8))) float v8f;

// Fast activations: v_exp_f32 + v_rcp_f32 (no IEEE divide sequences).
__device__ __forceinline__ float sigm_f(float x) {
    return __builtin_amdgcn_rcpf(1.0f + __expf(-x));
}
__device__ __forceinline__ float tanh_f(float x) {
    float e = __expf(2.0f * x);                  // e in (0, inf]
    return 1.0f - 2.0f * __builtin_amdgcn_rcpf(e + 1.0f);  // saturates to +/-1
}

// ---------------------------------------------------------------------------
// Kernel 1: per-timestep mask-sum normalizer  invden[t] = 1/(sum_b m[b,t]+1e-5)
// Also zeroes the global loss accumulator (ws[0]).
// ---------------------------------------------------------------------------
__global__ __launch_bounds__(256) void den_kernel(const float* __restrict__ masks,
                                                  float* __restrict__ ws) {
    __shared__ float red[256];
    const int t = blockIdx.x;
    float s = 0.0f;
    for (int b = threadIdx.x; b < B_; b += 256) s += masks[(size_t)b * T_ + t];
    red[threadIdx.x] = s;
    __syncthreads();
    for (int off = 128; off > 0; off >>= 1) {
        if ((int)threadIdx.x < off) red[threadIdx.x] += red[threadIdx.x + off];
        __syncthreads();
    }
    if (threadIdx.x == 0) {
        ws[1 + t] = 1.0f / (red[0] + 1e-5f);
        if (t == 0) ws[0] = 0.0f;
    }
}

// ---------------------------------------------------------------------------
// Kernel 2: pass-through copies (is_train, evals, eval_masks)
// ---------------------------------------------------------------------------
__global__ __launch_bounds__(256) void copy_kernel(const float* __restrict__ is_train,
                                                   const float* __restrict__ evals,
                                                   const float* __restrict__ eval_masks,
                                                   float* __restrict__ out) {
    size_t i = (size_t)blockIdx.x * blockDim.x + threadIdx.x;
    if (i < (size_t)B_) out[OFF_TRAIN + i] = is_train[i];
    if (i < NBT) {
        out[OFF_EVALS + i] = evals[i];
        out[OFF_EM + i]    = eval_masks[i];
    }
}

// ---------------------------------------------------------------------------
// Kernel 3: the sequential LSTM scan.
// One wave32 owns 16 batch rows. State h,c in WMMA C-layout fragments:
//   frag[r] at lane L  <->  (batch = r + 8*(L/16), hid = L%16)
// Per step (all in fp32 WMMA, 24 v_wmma per step):
//   x_h  = h(A) x Breg (w_reg broadcast over N), C = b_reg splat      [4 wmma]
//   gates= [h | x_c m 0 0](A, K=20) x [W_hh^T ; W_ih^T ; 0](B),
//          C = (b_ih+b_hh) splat                                      [20 wmma]
// ---------------------------------------------------------------------------
__global__ __launch_bounds__(128) void lstm_scan(const float* __restrict__ values,
                                                 const float* __restrict__ masks,
                                                 const float* __restrict__ W_ih,
                                                 const float* __restrict__ W_hh,
                                                 const float* __restrict__ b_ih,
                                                 const float* __restrict__ b_hh,
                                                 const float* __restrict__ W_reg,
                                                 const float* __restrict__ b_reg,
                                                 float* __restrict__ out_imp,
                                                 float* __restrict__ ws) {
    __shared__ float hsh[4][256];  // per-wave 16x16 f32 transpose tile

    const int lane = threadIdx.x & 31;
    const int wave = threadIdx.x >> 5;
    const int half = lane >> 4;    // 0: lanes 0-15, 1: lanes 16-31
    const int ln   = lane & 15;
    const int gb   = (blockIdx.x * 4 + wave) * 16;  // batch tile base
    float* hs = hsh[wave];
    const float* __restrict__ invden = ws + 1;

    // --- loop-invariant fragments ---
    // B of W_hh^T (K=hid x N=gate16) per gate tile nt, K-chunk kc:
    //   lane L: N = L%16 ; v0 <-> K = 4*kc + 2*(L/16), v1 <-> K+1
    v2f bfrag[4][5];     // kc=4 is the W_ih extension rows (K=16..19)
    v8f biasC[4];        // C-matrix splat of b_ih+b_hh per gate tile
#pragma unroll
    for (int nt = 0; nt < 4; ++nt) {
        const int g = nt * 16 + ln;  // global gate index (i,f,g,o order)
        const float bsum = b_ih[g] + b_hh[g];
#pragma unroll
        for (int e = 0; e < 8; ++e) biasC[nt][e] = bsum;
#pragma unroll
        for (int kc = 0; kc < 4; ++kc) {
            const int k = kc * 4 + half * 2;
            v2f bv;
            bv[0] = W_hh[g * 16 + k];
            bv[1] = W_hh[g * 16 + k + 1];
            bfrag[nt][kc] = bv;
        }
        // K=16 -> W_ih[g][0], K=17 -> W_ih[g][1], K=18,19 -> 0
        v2f be;
        be[0] = half ? 0.0f : W_ih[2 * g + 0];
        be[1] = half ? 0.0f : W_ih[2 * g + 1];
        bfrag[nt][4] = be;
    }
    // B of w_reg broadcast: Breg[k][n] = w_reg[k] for all n
    v2f bregf[4];
    v8f bregC;
#pragma unroll
    for (int kc = 0; kc < 4; ++kc) {
        const int k = kc * 4 + half * 2;
        v2f bv;
        bv[0] = W_reg[k];
        bv[1] = W_reg[k + 1];
        bregf[kc] = bv;
    }
    {
        const float br = b_reg[0];
#pragma unroll
        for (int e = 0; e < 8; ++e) bregC[e] = br;
    }

    const size_t rowbase = (size_t)(gb + ln) * T_;  // this lane's batch row

    float h[8], c[8];
#pragma unroll
    for (int r = 0; r < 8; ++r) { h[r] = 0.0f; c[r] = 0.0f; }
    float lossAcc = 0.0f;

    // software-pipelined input loads
    float x_l  = values[rowbase];
    float m_l  = masks[rowbase];
    float invd = invden[0];

    for (int t = 0; t < T_; ++t) {
        // ---- prefetch next step's inputs (overlaps with this step's compute)
        const int tn = (t + 1 < T_) ? (t + 1) : t;
        const float x_nx  = values[rowbase + tn];
        const float m_nx  = masks[rowbase + tn];
        const float iv_nx = invden[tn];

        // ---- transpose h (C-layout) -> A-layout via per-wave LDS tile
#pragma unroll
        for (int r = 0; r < 8; ++r) hs[(half * 8 + r) * 16 + ln] = h[r];
        asm volatile("s_wait_dscnt 0" ::: "memory");  // same-wave DS in-order
        v2f afrag[5];
#pragma unroll
        for (int kc = 0; kc < 4; ++kc)
            afrag[kc] = *(const v2f*)&hs[ln * 16 + kc * 4 + half * 2];

        // ---- x_h = h . w_reg + b_reg, broadcast over all N (lanes)
        v8f xacc = bregC;
#pragma unroll
        for (int kc = 0; kc < 4; ++kc)
            xacc = __builtin_amdgcn_wmma_f32_16x16x4_f32(
                false, afrag[kc], false, bregf[kc], (short)0, xacc, false, false);
        // extract x_h for this lane's batch (= ln): frag[r] holds batch r+8*half
        float own = xacc[0];
#pragma unroll
        for (int r = 1; r < 8; ++r)
            own = ((ln & 7) == r) ? xacc[r] : own;
        const float oth  = __shfl_xor(own, 16);
        const float xh_l = ((ln >> 3) == half) ? own : oth;

        // ---- imputation, loss, extended-A input column
        const float xc_l = m_l * x_l + (1.0f - m_l) * xh_l;
        lossAcc += fabsf(x_l - xh_l) * m_l * invd;  // halves duplicate; fixed at end
        if (half == 0) out_imp[rowbase + t] = xc_l;
        v2f a4;
        a4[0] = half ? 0.0f : xc_l;   // K=16 (half0) / K=18 (half1, zero-pad)
        a4[1] = half ? 0.0f : m_l;    // K=17 (half0) / K=19 (half1, zero-pad)
        afrag[4] = a4;

        // ---- gate GEMM: 4 gate tiles x 5 K-chunks; bias folded into C
        float gate[4][8];
#pragma unroll
        for (int nt = 0; nt < 4; ++nt) {
            v8f acc = biasC[nt];
#pragma unroll
            for (int kc = 0; kc < 5; ++kc)
                acc = __builtin_amdgcn_wmma_f32_16x16x4_f32(
                    false, afrag[kc], false, bfrag[nt][kc],
                    (short)0, acc, false, false);
#pragma unroll
            for (int r = 0; r < 8; ++r) gate[nt][r] = acc[r];
        }

        // ---- activations + cell update (frag element r <-> batch r+8*half)
#pragma unroll
        for (int r = 0; r < 8; ++r) {
            float I = sigm_f(gate[0][r]);
            float F = sigm_f(gate[1][r]);
            float G = tanh_f(gate[2][r]);
            float O = sigm_f(gate[3][r]);
            c[r] = F * c[r] + I * G;
            h[r] = O * tanh_f(c[r]);
        }

        x_l = x_nx; m_l = m_nx; invd = iv_nx;
    }

    // ---- final loss reduction: lanes L and L^16 hold identical per-batch sums
    lossAcc += __shfl_xor(lossAcc, 1);
    lossAcc += __shfl_xor(lossAcc, 2);
    lossAcc += __shfl_xor(lossAcc, 4);
    lossAcc += __shfl_xor(lossAcc, 8);
    lossAcc += __shfl_xor(lossAcc, 16);
    if (lane == 0) atomicAdd(&ws[0], 0.5f * lossAcc);
}

// ---------------------------------------------------------------------------
// Kernel 4: finalize loss
// ---------------------------------------------------------------------------
__global__ void fin_kernel(const float* __restrict__ ws, float* __restrict__ out) {
    out[0] = ws[0] / (float)T_;
}

extern "C" void kernel_launch(void* const* d_in, const int* in_sizes, int n_in,
                              void* d_out, int out_size, void* d_ws, size_t ws_size,
                              hipStream_t stream) {
    const float* values     = (const float*)d_in[0];
    const float* masks      = (const float*)d_in[1];
    const float* evals      = (const float*)d_in[2];
    const float* eval_masks = (const float*)d_in[3];
    const float* is_train   = (const float*)d_in[4];
    const float* W_ih       = (const float*)d_in[5];
    const float* W_hh       = (const float*)d_in[6];
    const float* b_ih       = (const float*)d_in[7];
    const float* b_hh       = (const float*)d_in[8];
    const float* W_reg      = (const float*)d_in[9];
    const float* b_reg      = (const float*)d_in[10];
    float* out = (float*)d_out;
    float* ws  = (float*)d_ws;   // ws[0]=loss acc, ws[1..T]=invden[t]

    (void)in_sizes; (void)n_in; (void)out_size; (void)ws_size;

    den_kernel<<<T_, 256, 0, stream>>>(masks, ws);
    copy_kernel<<<(int)((NBT + 255) / 256), 256, 0, stream>>>(is_train, evals, eval_masks, out);
    lstm_scan<<<B_ / 64, 128, 0, stream>>>(values, masks, W_ih, W_hh, b_ih, b_hh,
                                           W_reg, b_reg, out + OFF_IMP, ws);
    fin_kernel<<<1, 1, 0, stream>>>(ws, out);
}